// PositionEmbeddingLearned_42649025249307
// MI455X (gfx1250) — compile-verified
//
#include <hip/hip_runtime.h>

typedef __attribute__((ext_vector_type(2))) float v2f;
typedef __attribute__((ext_vector_type(8))) float v8f;

#define HID       128   // hidden dim (W1 cols / W2 rows)
#define H_OUT     256   // output dim H
#define T_O       32
#define N_MAX     24
#define ROW_TILE  64    // rows per block: 4 WMMA M-tiles -> 4x reuse of B frags
#define SH_STRIDE 130   // 128 + 2 pad -> conflict-free ds_load_b64 on A fragments

// ---------------------------------------------------------------------------
// Kernel 1: objbase[obj] = slot*(B*T_O) + frame*T_O   (ragged -> padded map)
// ---------------------------------------------------------------------------
__global__ void build_table_kernel(const int* __restrict__ n_per_frame,
                                   int* __restrict__ objbase, int B) {
    __shared__ int starts[512];
    const int t = threadIdx.x;
    if (t == 0) {                    // B == 256: trivial sequential scan
        int acc = 0;
        for (int b = 0; b < B; ++b) { starts[b] = acc; acc += n_per_frame[b]; }
    }
    __syncthreads();
    for (int b = t; b < B; b += blockDim.x) {
        const int s0 = starts[b];
        const int nb = n_per_frame[b];
        for (int s = 0; s < nb; ++s)
            objbase[s0 + s] = s * (B * T_O) + b * T_O;
    }
}

// ---------------------------------------------------------------------------
// Kernel 2: zero only the INVALID (frame, slot) tiles: T_O*H_OUT floats each.
// Total zero traffic ~51 MB instead of memsetting the full 201 MB output.
// ---------------------------------------------------------------------------
__global__ void zero_pad_kernel(const int* __restrict__ n_per_frame,
                                float* __restrict__ out, int B) {
    const int id   = blockIdx.x;
    const int b    = id / N_MAX;
    const int slot = id % N_MAX;
    if (slot < n_per_frame[b]) return;
    float4* p = (float4*)(out + ((size_t)slot * B * T_O + (size_t)b * T_O) * H_OUT);
    const float4 z = make_float4(0.f, 0.f, 0.f, 0.f);
    for (int i = threadIdx.x; i < (T_O * H_OUT) / 4; i += blockDim.x)
        p[i] = z;
}

// ---------------------------------------------------------------------------
// Kernel 3: fused MLP + scatter.
// Block = 256 threads (8 wave32), handles 64 rows x 256 cols.
//  - GEMM1 (K=4) on VALU into LDS h[64][130] (relu applied)
//  - GEMM2 via v_wmma_f32_16x16x4_f32: per wave 2 column tiles x 4 row tiles;
//    each B fragment (2 global b32 loads) feeds 4 WMMAs -> 4x less W2/L2
//    traffic than a 16-row tile.
//  - epilogue: +b2, scatter-store each row via rowbase[] table
// No divergent control flow -> EXEC all-1s at every WMMA.
// ---------------------------------------------------------------------------
__global__ void __launch_bounds__(256)
mlp_scatter_wmma_kernel(const float* __restrict__ bbox,
                        const float* __restrict__ W1,
                        const float* __restrict__ b1,
                        const float* __restrict__ W2,
                        const float* __restrict__ b2,
                        const int*  __restrict__ objbase,
                        float* __restrict__ out) {
    __shared__ float sh_h[ROW_TILE * SH_STRIDE];
    __shared__ int   sh_rowbase[ROW_TILE];

    const int t    = threadIdx.x;
    const int row0 = blockIdx.x * ROW_TILE;

    // per-row scattered output base index (fits in int: 24*8192*256 < 2^31)
    if (t < ROW_TILE) {
        const int r   = row0 + t;
        const int obj = r / T_O;
        const int tt  = r % T_O;
        sh_rowbase[t] = (objbase[obj] + tt) * H_OUT;
    }

    // ---- GEMM1: h = relu(bbox @ W1 + b1). Each thread: 4 rows x 8 cols. ----
    {
        const int mr = t >> 4;                 // 0..15
        const int j0 = (t & 15) * (HID / 16);  // 8 hidden outputs per row
        float4 bb[4];
        #pragma unroll
        for (int q = 0; q < 4; ++q)
            bb[q] = ((const float4*)bbox)[row0 + mr + q * 16];
        #pragma unroll
        for (int jj = 0; jj < HID / 16; ++jj) {
            const int j = j0 + jj;
            const float w0 = W1[0 * HID + j];
            const float w1 = W1[1 * HID + j];
            const float w2 = W1[2 * HID + j];
            const float w3 = W1[3 * HID + j];
            const float bias = b1[j];
            #pragma unroll
            for (int q = 0; q < 4; ++q) {
                float a = bias;
                a = fmaf(bb[q].x, w0, a);
                a = fmaf(bb[q].y, w1, a);
                a = fmaf(bb[q].z, w2, a);
                a = fmaf(bb[q].w, w3, a);
                sh_h[(mr + q * 16) * SH_STRIDE + j] = fmaxf(a, 0.0f);
            }
        }
    }
    __syncthreads();

    // ---- GEMM2 via f32 WMMA ----
    const int lane = t & 31;
    const int wave = t >> 5;
    const int hi   = lane >> 4;   // half-wave: selects K{0,1} vs K{2,3}
    const int ln   = lane & 15;   // M (for A) / N (for B)

    v8f acc[4][2] = {};              // [row tile][column tile]
    const int c0 = wave * 32 + ln;   // column of tile u=0
    const int c1 = c0 + 16;          // column of tile u=1

    #pragma unroll 2
    for (int k = 0; k < HID / 4; ++k) {
        // B fragments: lane holds W2[4k + 2*hi + {0,1}][col]; reused 4x below
        const int r0 = (4 * k + 2 * hi) * H_OUT;
        v2f bf0, bf1;
        bf0.x = W2[r0 + c0];
        bf0.y = W2[r0 + H_OUT + c0];
        bf1.x = W2[r0 + c1];
        bf1.y = W2[r0 + H_OUT + c1];
        #pragma unroll
        for (int mt = 0; mt < 4; ++mt) {
            // A fragment: lane ln holds h[mt*16 + ln][4k + 2*hi + {0,1}]
            const v2f a =
                *(const v2f*)&sh_h[(mt * 16 + ln) * SH_STRIDE + 4 * k + 2 * hi];
            acc[mt][0] = __builtin_amdgcn_wmma_f32_16x16x4_f32(
                             false, a, false, bf0, (short)0, acc[mt][0], false, false);
            acc[mt][1] = __builtin_amdgcn_wmma_f32_16x16x4_f32(
                             false, a, false, bf1, (short)0, acc[mt][1], false, false);
        }
    }

    // ---- epilogue: bias + scatter. C/D layout: VGPR v holds row v + 8*hi ----
    const float bias0 = b2[c0];
    const float bias1 = b2[c1];
    #pragma unroll
    for (int mt = 0; mt < 4; ++mt) {
        #pragma unroll
        for (int v = 0; v < 8; ++v) {
            const int rb = sh_rowbase[mt * 16 + v + 8 * hi];
            out[rb + c0] = acc[mt][0][v] + bias0;
            out[rb + c1] = acc[mt][1][v] + bias1;
        }
    }
}

// ---------------------------------------------------------------------------
// Inputs (setup_inputs order): 0=bbox f32, 1=n_per_frame i32, 2=n_max,
// 3=T_o, 4=W1, 5=b1, 6=W2, 7=b2.  Output: f32 [24, 8192, 256].
// ---------------------------------------------------------------------------
extern "C" void kernel_launch(void* const* d_in, const int* in_sizes, int n_in,
                              void* d_out, int out_size, void* d_ws, size_t ws_size,
                              hipStream_t stream) {
    const float* bbox = (const float*)d_in[0];
    const int*   npf  = (const int*)  d_in[1];
    const float* W1   = (const float*)d_in[4];
    const float* b1   = (const float*)d_in[5];
    const float* W2   = (const float*)d_in[6];
    const float* b2   = (const float*)d_in[7];
    float* out = (float*)d_out;

    const int B  = in_sizes[1];          // 256
    const int NT = in_sizes[0] / 4;      // N_total * T_o = 146880 (multiple of 64)

    int* objbase = (int*)d_ws;           // N_total ints (~18 KB) in scratch

    build_table_kernel<<<1, 256, 0, stream>>>(npf, objbase, B);
    zero_pad_kernel<<<B * N_MAX, 256, 0, stream>>>(npf, out, B);
    mlp_scatter_wmma_kernel<<<NT / ROW_TILE, 256, 0, stream>>>(
        bbox, W1, b1, W2, b2, objbase, out);
}